// GINConv_2345052143748
// MI455X (gfx1250) — compile-verified
//
#include <hip/hip_runtime.h>

#define D_FEAT 128
#define TILE   1024
#define BLOCK  256

// 16-byte int vector type expected by the async global->LDS b128 builtin
typedef int v4i __attribute__((ext_vector_type(4)));
typedef __attribute__((address_space(1))) v4i gv4i_t;   // global (AS1)
typedef __attribute__((address_space(3))) v4i lv4i_t;   // LDS (AS3)

// Pass 1: out = (1+eps) * feat, vectorized float4 -> global_load_b128 / global_store_b128
__global__ void __launch_bounds__(BLOCK) gin_init(const float4* __restrict__ feat4,
                                                  const float* __restrict__ eps,
                                                  float4* __restrict__ out4,
                                                  int n4) {
    int i = blockIdx.x * BLOCK + threadIdx.x;
    if (i >= n4) return;
    float s = 1.0f + eps[0];
    float4 f = feat4[i];
    float4 r;
    r.x = s * f.x;
    r.y = s * f.y;
    r.z = s * f.z;
    r.w = s * f.w;
    out4[i] = r;
}

__device__ __forceinline__ void edge_accum(const float* __restrict__ feat,
                                           float* __restrict__ out,
                                           int s, int d, int lane) {
    // s,d are wave-uniform SGPR values -> scalar base address + per-lane voffset
    const float4 v = ((const float4*)(feat + (size_t)(unsigned)s * D_FEAT))[lane];
    float* o = out + (size_t)(unsigned)d * D_FEAT + (size_t)lane * 4;
    __hip_atomic_fetch_add(&o[0], v.x, __ATOMIC_RELAXED, __HIP_MEMORY_SCOPE_AGENT);
    __hip_atomic_fetch_add(&o[1], v.y, __ATOMIC_RELAXED, __HIP_MEMORY_SCOPE_AGENT);
    __hip_atomic_fetch_add(&o[2], v.z, __ATOMIC_RELAXED, __HIP_MEMORY_SCOPE_AGENT);
    __hip_atomic_fetch_add(&o[3], v.w, __ATOMIC_RELAXED, __HIP_MEMORY_SCOPE_AGENT);
}

// Pass 2: per-edge scatter-add. Block stages a TILE of int64 src/dst indices into LDS
// via CDNA5 async global->LDS B128 copies; 8 waves then process one edge each per step:
// wave-uniform indices forced into SGPRs (readfirstlane), one float4 (b128) gather per
// lane from the L2-resident feature table, 4 fp32 atomic adds per lane into out[dst].
__global__ void __launch_bounds__(BLOCK) gin_scatter(const float* __restrict__ feat,
                                                     const long long* __restrict__ src,
                                                     const long long* __restrict__ dst,
                                                     float* __restrict__ out,
                                                     int n_edges) {
    __shared__ long long lds_src[TILE];
    __shared__ long long lds_dst[TILE];

    const int tbase  = blockIdx.x * TILE;
    int tcount = n_edges - tbase;
    if (tcount > TILE) tcount = TILE;
    const int t = threadIdx.x;

#if __has_builtin(__builtin_amdgcn_global_load_async_to_lds_b128) && \
    __has_builtin(__builtin_amdgcn_s_wait_asynccnt)
    if (tcount == TILE) {
        gv4i_t* gs = (gv4i_t*)(src + tbase);   // TILE int64 = TILE/2 b128 chunks
        gv4i_t* gd = (gv4i_t*)(dst + tbase);
        lv4i_t* ls = (lv4i_t*)lds_src;
        lv4i_t* ld = (lv4i_t*)lds_dst;
        for (int c = t; c < TILE / 2; c += BLOCK) {
            __builtin_amdgcn_global_load_async_to_lds_b128(gs + c, ls + c, 0, 0);
            __builtin_amdgcn_global_load_async_to_lds_b128(gd + c, ld + c, 0, 0);
        }
        __builtin_amdgcn_s_wait_asynccnt(0);
    } else
#endif
    {
        for (int i = t; i < tcount; i += BLOCK) {
            lds_src[i] = src[tbase + i];
            lds_dst[i] = dst[tbase + i];
        }
    }
    __syncthreads();

    const int lane = t & 31;
    const int wid  = t >> 5;
    const int nw   = BLOCK / 32;

    // Low 32 bits of the int64 indices (node ids < 2^31); wave-uniform LDS reads.
    const int* lsrc32 = (const int*)lds_src;
    const int* ldst32 = (const int*)lds_dst;

    int i = wid;
    // Unroll by 2: issue both gathers before the atomics to overlap L2 latency.
    for (; i + nw < tcount; i += 2 * nw) {
        int s0 = __builtin_amdgcn_readfirstlane(lsrc32[2 * i]);
        int d0 = __builtin_amdgcn_readfirstlane(ldst32[2 * i]);
        int s1 = __builtin_amdgcn_readfirstlane(lsrc32[2 * (i + nw)]);
        int d1 = __builtin_amdgcn_readfirstlane(ldst32[2 * (i + nw)]);

        const float4 v0 = ((const float4*)(feat + (size_t)(unsigned)s0 * D_FEAT))[lane];
        const float4 v1 = ((const float4*)(feat + (size_t)(unsigned)s1 * D_FEAT))[lane];
        float* o0 = out + (size_t)(unsigned)d0 * D_FEAT + (size_t)lane * 4;
        float* o1 = out + (size_t)(unsigned)d1 * D_FEAT + (size_t)lane * 4;

        __hip_atomic_fetch_add(&o0[0], v0.x, __ATOMIC_RELAXED, __HIP_MEMORY_SCOPE_AGENT);
        __hip_atomic_fetch_add(&o0[1], v0.y, __ATOMIC_RELAXED, __HIP_MEMORY_SCOPE_AGENT);
        __hip_atomic_fetch_add(&o0[2], v0.z, __ATOMIC_RELAXED, __HIP_MEMORY_SCOPE_AGENT);
        __hip_atomic_fetch_add(&o0[3], v0.w, __ATOMIC_RELAXED, __HIP_MEMORY_SCOPE_AGENT);
        __hip_atomic_fetch_add(&o1[0], v1.x, __ATOMIC_RELAXED, __HIP_MEMORY_SCOPE_AGENT);
        __hip_atomic_fetch_add(&o1[1], v1.y, __ATOMIC_RELAXED, __HIP_MEMORY_SCOPE_AGENT);
        __hip_atomic_fetch_add(&o1[2], v1.z, __ATOMIC_RELAXED, __HIP_MEMORY_SCOPE_AGENT);
        __hip_atomic_fetch_add(&o1[3], v1.w, __ATOMIC_RELAXED, __HIP_MEMORY_SCOPE_AGENT);
    }
    if (i < tcount) {
        int s0 = __builtin_amdgcn_readfirstlane(lsrc32[2 * i]);
        int d0 = __builtin_amdgcn_readfirstlane(ldst32[2 * i]);
        edge_accum(feat, out, s0, d0, lane);
    }
}

extern "C" void kernel_launch(void* const* d_in, const int* in_sizes, int n_in,
                              void* d_out, int out_size, void* d_ws, size_t ws_size,
                              hipStream_t stream) {
    const float*     feat = (const float*)d_in[0];       // [N, 128] f32
    const float*     eps  = (const float*)d_in[1];       // [1] f32
    const long long* src  = (const long long*)d_in[2];   // [E] i64
    const long long* dst  = (const long long*)d_in[3];   // [E] i64
    float*           out  = (float*)d_out;               // [N, 128] f32

    const int nfeat   = in_sizes[0];      // N * 128
    const int n4      = nfeat / 4;        // float4 count
    const int n_edges = in_sizes[2];

    // Pass 1: out = (1+eps) * feat
    gin_init<<<(n4 + BLOCK - 1) / BLOCK, BLOCK, 0, stream>>>(
        (const float4*)feat, eps, (float4*)out, n4);

    // Pass 2: scatter-add neighbor features (same stream -> ordered after init)
    gin_scatter<<<(n_edges + TILE - 1) / TILE, BLOCK, 0, stream>>>(
        feat, src, dst, out, n_edges);
}